// TranAttLayer_68977174774383
// MI455X (gfx1250) — compile-verified
//
#include <hip/hip_runtime.h>

typedef __attribute__((ext_vector_type(2))) float v2f;
typedef __attribute__((ext_vector_type(8))) float v8f;

#define NEG_SLOPE 0.01f

// ---------------------------------------------------------------------------
// Kernel 1: fold Wa into the skinny GEMM operand.
//   u_l[k] = sum_e Wa[e,k] * w_l[e]     (since l = (X @ Wa^T) @ w_l = X @ (Wa^T w_l))
//   u_r[k] = sum_e Wa[e,k] * w_r[e]
// Build U[128][16] row-major: col0 = u_l, col1 = u_r, col2 = W_demand, rest 0.
// ---------------------------------------------------------------------------
__global__ void build_U_kernel(const float* __restrict__ Wa,
                               const float* __restrict__ w_l,
                               const float* __restrict__ w_r,
                               const float* __restrict__ W_demand,
                               float* __restrict__ U, int D) {
  int k = threadIdx.x;            // 0..D-1
  float ul = 0.f, ur = 0.f;
  for (int e = 0; e < D; ++e) {
    float w = Wa[e * D + k];      // coalesced across threads
    ul += w * w_l[e];
    ur += w * w_r[e];
  }
#pragma unroll
  for (int n = 0; n < 16; ++n) U[k * 16 + n] = 0.f;
  U[k * 16 + 0] = ul;
  U[k * 16 + 1] = ur;
  U[k * 16 + 2] = W_demand[k];
}

// ---------------------------------------------------------------------------
// Kernel 2: l / r / demands via V_WMMA_F32_16X16X4_F32.
// One wave (32 lanes) per 16-row tile of X[R][128]: C[16x16] = sum_k A(16x4)*B(4x16).
// f32 A 16x4 layout: lane<16 -> M=lane, {v0,v1}={K0,K1}; lane>=16 -> {K2,K3}.
// f32 B 4x16 layout (mirror): lane<16 -> N=lane, {v0,v1}={K0,K1}; lane>=16 -> {K2,K3}.
// C/D layout: value(lane, vgpr v) = C[M = v + 8*(lane>>4)][N = lane&15].
// ---------------------------------------------------------------------------
__global__ __launch_bounds__(32) void lrd_wmma_kernel(
    const float* __restrict__ X, const float* __restrict__ U,
    const float* __restrict__ b_demand,
    float* __restrict__ l_ws, float* __restrict__ r_ws,
    float* __restrict__ dem_ws, float* __restrict__ dem_out, int D) {
  const int lane = threadIdx.x;
  const int m = lane & 15;
  const int h = lane >> 4;                 // 0 or 1: which K pair
  const long row0 = (long)blockIdx.x * 16;

  const float* arow = X + (row0 + m) * D + 2 * h;  // this lane's A row slice
  const float* bcol = U + (2 * h) * 16 + m;        // this lane's B column slice

  v8f c = {0.f, 0.f, 0.f, 0.f, 0.f, 0.f, 0.f, 0.f};

#pragma unroll
  for (int kk = 0; kk < 32; ++kk) {        // K = 128 = 32 steps of 4
    const int k0 = kk * 4;
    const float2 av = *(const float2*)(arow + k0);   // 8B-aligned (k0+2h even)
    v2f a; a.x = av.x; a.y = av.y;
    v2f b; b.x = bcol[k0 * 16]; b.y = bcol[k0 * 16 + 16];
    c = __builtin_amdgcn_wmma_f32_16x16x4_f32(
        /*neg_a=*/false, a, /*neg_b=*/false, b,
        /*c_mod=*/(short)0, c, /*reuse_a=*/false, /*reuse_b=*/false);
  }

  const float bd = b_demand[0];
#pragma unroll
  for (int v = 0; v < 8; ++v) {
    const long row = row0 + v + 8 * h;     // global row = M index in tile
    const float val = c[v];
    if (m == 0) {                          // column 0 = l
      l_ws[row] = val;
    } else if (m == 1) {                   // column 1 = r
      r_ws[row] = val;
    } else if (m == 2) {                   // column 2 = demand logit
      const float d = 1.0f / (1.0f + __expf(-(val + bd)));
      dem_ws[row] = d;
      dem_out[row] = d;                    // demands output (first tuple element)
    }
  }
}

// ---------------------------------------------------------------------------
// Kernel 3: per-row softmax + demand scaling (memory-bound: 134 MB of stores).
// One 256-thread workgroup (8 wave32) per output row i of batch b:
//   a_j = leaky_relu(l_i + r_j)  -> LDS;  amax;  e_j = exp(a_j - amax);  sum;
//   Gs[b,i,j] = e_j * (demand_i / sum)
// r[b,:] is 16 KB -> stays resident in L2/WGP$, so HBM traffic ~= output only.
// ---------------------------------------------------------------------------
__global__ __launch_bounds__(256) void softmax_row_kernel(
    const float* __restrict__ l_ws, const float* __restrict__ r_ws,
    const float* __restrict__ dem_ws, float* __restrict__ Gs, int N) {
  extern __shared__ float smem[];          // N floats for a_j/e_j + 8 partials
  float* sA = smem;
  float* sRed = smem + N;

  const int row = blockIdx.x;
  const int b = row / N;
  const int tid = threadIdx.x;
  const int wid = tid >> 5;

  const float li = l_ws[row];
  const float* rb = r_ws + (long)b * N;

  // pass 1: leaky_relu into LDS, track max
  float lmax = -3.402823466e38f;
  for (int j = tid; j < N; j += 256) {
    const float t = li + rb[j];
    const float a = t > 0.f ? t : NEG_SLOPE * t;
    sA[j] = a;
    lmax = fmaxf(lmax, a);
  }
#pragma unroll
  for (int off = 16; off > 0; off >>= 1)
    lmax = fmaxf(lmax, __shfl_xor(lmax, off, 32));
  if ((tid & 31) == 0) sRed[wid] = lmax;
  __syncthreads();
  float amax = sRed[0];
#pragma unroll
  for (int k = 1; k < 8; ++k) amax = fmaxf(amax, sRed[k]);

  // pass 2: exp into LDS, track sum
  float lsum = 0.f;
  for (int j = tid; j < N; j += 256) {
    const float e = __expf(sA[j] - amax);
    sA[j] = e;
    lsum += e;
  }
#pragma unroll
  for (int off = 16; off > 0; off >>= 1)
    lsum += __shfl_xor(lsum, off, 32);
  __syncthreads();                         // all amax reads done before reuse
  if ((tid & 31) == 0) sRed[wid] = lsum;
  __syncthreads();
  float sum = 0.f;
#pragma unroll
  for (int k = 0; k < 8; ++k) sum += sRed[k];

  const float scale = dem_ws[row] / sum;

  // pass 3: coalesced normalized stores
  float* out = Gs + (size_t)row * N;
  for (int j = tid; j < N; j += 256) out[j] = sA[j] * scale;
}

// ---------------------------------------------------------------------------
// Launch. Inputs: 0=embed_feat[B,N,D] 1=predict_G 2=W_demand[D] 3=b_demand[1]
//                 4=Wa[D,D] 5=w_l[D] 6=w_r[D]
// Output: demands[B*N] then Gs[B*N*N], concatenated flat.
// ---------------------------------------------------------------------------
extern "C" void kernel_launch(void* const* d_in, const int* in_sizes, int n_in,
                              void* d_out, int out_size, void* d_ws, size_t ws_size,
                              hipStream_t stream) {
  if (n_in < 7) return;
  const float* X        = (const float*)d_in[0];
  const float* W_demand = (const float*)d_in[2];
  const float* b_demand = (const float*)d_in[3];
  const float* Wa       = (const float*)d_in[4];
  const float* w_l      = (const float*)d_in[5];
  const float* w_r      = (const float*)d_in[6];

  const int D = in_sizes[2];               // 128
  const int R = in_sizes[0] / D;           // B*N total rows (8192)
  const int N = out_size / R - 1;          // out = R + R*N  ->  N = 4096
  (void)ws_size;

  float* ws      = (float*)d_ws;
  float* U       = ws;                     // D*16 floats
  float* l_ws    = U + (size_t)D * 16;     // R floats
  float* r_ws    = l_ws + R;               // R floats
  float* dem_ws  = r_ws + R;               // R floats

  float* dem_out = (float*)d_out;          // [R]
  float* Gs_out  = dem_out + R;            // [R*N]

  build_U_kernel<<<1, D, 0, stream>>>(Wa, w_l, w_r, W_demand, U, D);
  lrd_wmma_kernel<<<R / 16, 32, 0, stream>>>(X, U, b_demand, l_ws, r_ws,
                                             dem_ws, dem_out, D);
  softmax_row_kernel<<<R, 256, (N + 8) * sizeof(float), stream>>>(
      l_ws, r_ws, dem_ws, Gs_out, N);
}